// HigherPointPart_68710886801960
// MI455X (gfx1250) — compile-verified
//
#include <hip/hip_runtime.h>

typedef float v2f __attribute__((ext_vector_type(2)));
typedef float v8f __attribute__((ext_vector_type(8)));

#define N_CONST 100000
#define B_SEG   1024
#define EPS_BN  1e-5f

// ---------------------------------------------------------------------------
// Jet accumulation: jet[ind[i], c] += x[i,c] * w[i]   (segment sums for BN)
// One thread per (node, 4-channel group): float4 load + 4 f32 atomics.
// ---------------------------------------------------------------------------
__global__ void jet_kernel(const float* __restrict__ X, const float* __restrict__ w,
                           const int* __restrict__ ind, float* __restrict__ jet,
                           int n, int C) {
    int per = C >> 2;
    int t = blockIdx.x * blockDim.x + threadIdx.x;
    if (t >= n * per) return;
    int node = t / per;
    int q    = t - node * per;
    float4 x = ((const float4*)X)[(size_t)node * per + q];
    float ww = w[node];
    int b = ind[node];
    float* dst = jet + (size_t)b * C + q * 4;
    atomicAdd(dst + 0, x.x * ww);
    atomicAdd(dst + 1, x.y * ww);
    atomicAdd(dst + 2, x.z * ww);
    atomicAdd(dst + 3, x.w * ww);
}

// ---------------------------------------------------------------------------
// BN stats: mean/var (ddof=1) over the B_SEG segment sums; fold gamma/beta
// into per-channel scale/shift:  xn = x*scale + shift
// ---------------------------------------------------------------------------
__global__ void bnstats_kernel(const float* __restrict__ jet,
                               const float* __restrict__ g, const float* __restrict__ b,
                               float* __restrict__ scale, float* __restrict__ shift, int C) {
    int c = threadIdx.x;
    if (c >= C) return;
    float s = 0.f, s2 = 0.f;
    for (int i = 0; i < B_SEG; ++i) {
        float v = jet[(size_t)i * C + c];
        s += v; s2 += v * v;
    }
    float mean = s * (1.f / B_SEG);
    float var  = (s2 - (float)B_SEG * mean * mean) * (1.f / (B_SEG - 1));
    float inv  = rsqrtf(var + EPS_BN);
    float sc   = g[c] * inv;
    scale[c] = sc;
    shift[c] = b[c] - mean * sc;
}

// ---------------------------------------------------------------------------
// Fused BN(+ReLU)(+row-weight) + GEMM via V_WMMA_F32_16X16X4_F32.
//   out[i,j] (+)= act( A[i,:] )  @  W[j,:]        W is (OUTW, K) row-major
// Block = 128 threads (4 waves), 64 rows per block, K chunked by 32 in LDS.
// A fragment: lane<16 holds row M=l2, K=k0..k0+1; lane>=16 K=k0+2..k0+3.
// B fragment: VGPR r, K = k0 + 2*hi + r, N = l2.   C/D: row = r + 8*hi, col = l2.
// ---------------------------------------------------------------------------
template <int K, int OUTW, bool RELU, bool ADDOUT, bool NORM, bool WMUL>
__global__ __launch_bounds__(128) void gemm_bn_kernel(
    const float* __restrict__ A, const float* __restrict__ scale,
    const float* __restrict__ shift, const float* __restrict__ W,
    const float* __restrict__ wmul, float* __restrict__ out, int nrows) {

    constexpr int NT = OUTW / 16;          // output 16-col tiles
    __shared__ float sA[4 * 16 * (K + 1)]; // 4 waves x 16 rows, padded
    __shared__ float sB[32 * (OUTW + 1)];  // one K-chunk of W^T, padded

    const int rowBase = blockIdx.x * 64;
    const int lane = threadIdx.x & 31;
    const int wave = threadIdx.x >> 5;
    const int l2 = lane & 15;
    const int hi = lane >> 4;

    // Stage A tile: normalize + relu + per-row weight, zeros past nrows.
    for (int idx = threadIdx.x; idx < 64 * K; idx += 128) {
        int r = idx / K, c = idx - r * K;
        int row = rowBase + r;
        float v = 0.f;
        if (row < nrows) {
            v = A[(size_t)row * K + c];
            if (NORM) v = v * scale[c] + shift[c];
            if (RELU) v = fmaxf(v, 0.f);
            if (WMUL) v *= wmul[row];
        }
        sA[r * (K + 1) + c] = v;
    }

    const int myRow = rowBase + wave * 16;
    v8f acc[NT];
    for (int jt = 0; jt < NT; ++jt) {
        for (int r = 0; r < 8; ++r) {
            int row = myRow + r + 8 * hi;
            float c0 = 0.f;
            if (ADDOUT && row < nrows) c0 = out[(size_t)row * OUTW + jt * 16 + l2];
            acc[jt][r] = c0;
        }
    }

    const float* sAw = &sA[wave * 16 * (K + 1)];
    for (int kc = 0; kc < K; kc += 32) {
        __syncthreads();
        // Stage B chunk transposed: sB[c][j] = W[j][kc+c]; coalesced on c.
        for (int idx = threadIdx.x; idx < 32 * OUTW; idx += 128) {
            int c = idx & 31, j = idx >> 5;
            sB[c * (OUTW + 1) + j] = W[(size_t)j * K + kc + c];
        }
        __syncthreads();
        for (int k0 = 0; k0 < 32; k0 += 4) {
            int ka = k0 + 2 * hi;
            v2f af;
            af.x = sAw[l2 * (K + 1) + kc + ka];
            af.y = sAw[l2 * (K + 1) + kc + ka + 1];
            for (int jt = 0; jt < NT; ++jt) {
                v2f bf;
                bf.x = sB[ka       * (OUTW + 1) + jt * 16 + l2];
                bf.y = sB[(ka + 1) * (OUTW + 1) + jt * 16 + l2];
                acc[jt] = __builtin_amdgcn_wmma_f32_16x16x4_f32(
                    false, af, false, bf, (short)0, acc[jt], false, false);
            }
        }
    }

    for (int jt = 0; jt < NT; ++jt)
        for (int r = 0; r < 8; ++r) {
            int row = myRow + r + 8 * hi;
            if (row < nrows) out[(size_t)row * OUTW + jt * 16 + l2] = acc[jt][r];
        }
}

// ---------------------------------------------------------------------------
// CSR build: degree histogram -> block scan -> cursor copy -> scatter perm
// ---------------------------------------------------------------------------
__global__ void deg_kernel(const int* __restrict__ head, int* __restrict__ deg, int E) {
    int e = blockIdx.x * blockDim.x + threadIdx.x;
    if (e < E) atomicAdd(&deg[head[e]], 1);
}

__global__ void scan_kernel(const int* __restrict__ deg, int* __restrict__ off, int n) {
    __shared__ int part[256];
    int tid = threadIdx.x;
    int chunk = (n + 255) / 256;
    int beg = tid * chunk;
    int end = min(beg + chunk, n);
    int s = 0;
    for (int i = beg; i < end; ++i) s += deg[i];
    part[tid] = s;
    __syncthreads();
    if (tid == 0) {
        int run = 0;
        for (int i = 0; i < 256; ++i) { int v = part[i]; part[i] = run; run += v; }
    }
    __syncthreads();
    int run = part[tid];
    for (int i = beg; i < end; ++i) { off[i] = run; run += deg[i]; }
    if (tid == 255) off[n] = run;
}

__global__ void copy_kernel(const int* __restrict__ src, int* __restrict__ dst, int n) {
    int i = blockIdx.x * blockDim.x + threadIdx.x;
    if (i < n) dst[i] = src[i];
}

__global__ void scatter_kernel(const int* __restrict__ head, int* __restrict__ cur,
                               int* __restrict__ perm, int E) {
    int e = blockIdx.x * blockDim.x + threadIdx.x;
    if (e < E) {
        int pos = atomicAdd(&cur[head[e]], 1);
        perm[pos] = e;
    }
}

// ---------------------------------------------------------------------------
// Gather aggregation: agg[n, j*4+t] = sum_{e: head==n} h1[tail[e], j] * pf[e,t]
// One wave per node; lane = channel j (0..31); 4 accumulators per lane.
// Replaces 205M fp32 atomics with streaming L2 reads + one 512B row store.
// ---------------------------------------------------------------------------
__global__ __launch_bounds__(256) void gather_kernel(
    const float* __restrict__ h1, const float4* __restrict__ pf,
    const int* __restrict__ tail, const int* __restrict__ off,
    const int* __restrict__ perm, float4* __restrict__ agg4, int n) {
    int wave = threadIdx.x >> 5;
    int lane = threadIdx.x & 31;
    int node = blockIdx.x * 8 + wave;
    if (node >= n) return;
    int beg = off[node], end = off[node + 1];
    float a0 = 0.f, a1 = 0.f, a2 = 0.f, a3 = 0.f;
    for (int idx = beg; idx < end; ++idx) {
        int e = perm[idx];
        int t = tail[e];
        float hv = h1[(size_t)t * 32 + lane];
        float4 p = pf[e];
        a0 = fmaf(hv, p.x, a0);
        a1 = fmaf(hv, p.y, a1);
        a2 = fmaf(hv, p.z, a2);
        a3 = fmaf(hv, p.w, a3);
    }
    agg4[(size_t)node * 32 + lane] = make_float4(a0, a1, a2, a3);
}

// ---------------------------------------------------------------------------
extern "C" void kernel_launch(void* const* d_in, const int* in_sizes, int n_in,
                              void* d_out, int out_size, void* d_ws, size_t ws_size,
                              hipStream_t stream) {
    const float* x    = (const float*)d_in[0];
    const float* w    = (const float*)d_in[1];
    const float* pf   = (const float*)d_in[2];
    const int*   head = (const int*)d_in[3];
    const int*   tail = (const int*)d_in[4];
    const int*   ind  = (const int*)d_in[5];
    const float* Wsc  = (const float*)d_in[6];
    const float* g1   = (const float*)d_in[7];
    const float* b1   = (const float*)d_in[8];
    const float* W1   = (const float*)d_in[9];
    const float* g2   = (const float*)d_in[10];
    const float* b2   = (const float*)d_in[11];
    const float* W2   = (const float*)d_in[12];
    const float* g3   = (const float*)d_in[13];
    const float* b3   = (const float*)d_in[14];
    const float* W3   = (const float*)d_in[15];
    float* out = (float*)d_out;

    const int N = in_sizes[1];          // part_weight has N elements
    const int E = in_sizes[3];          // pair_head has E elements

    // Workspace layout (256B aligned slices); ~72 MB total.
    char* p = (char*)d_ws;
    auto carve = [&](size_t bytes) -> char* {
        char* r = p; p += (bytes + 255) & ~(size_t)255; return r;
    };
    float* h1   = (float*)carve((size_t)N * 32 * 4);
    float* agg  = (float*)carve((size_t)N * 128 * 4);   // also holds h2 in-place
    float* jet  = (float*)carve((size_t)B_SEG * 128 * 4);
    float* st   = (float*)carve(6 * 128 * 4);
    int*   deg  = (int*)carve((size_t)N * 4);
    int*   off  = (int*)carve((size_t)(N + 1) * 4);
    int*   cur  = (int*)carve((size_t)N * 4);
    int*   perm = (int*)carve((size_t)E * 4);
    float *scale1 = st,       *shift1 = st + 128;
    float *scale2 = st + 256, *shift2 = st + 384;
    float *scale3 = st + 512, *shift3 = st + 640;

    hipMemsetAsync(jet, 0, (size_t)B_SEG * 128 * 4, stream);
    hipMemsetAsync(deg, 0, (size_t)N * 4, stream);

    // --- BN1 stats + stage-1 GEMM: h1 = relu(bn1(x)) @ W1^T * w  (N x 32)
    jet_kernel<<<(N * 16 + 255) / 256, 256, 0, stream>>>(x, w, ind, jet, N, 64);
    bnstats_kernel<<<1, 64, 0, stream>>>(jet, g1, b1, scale1, shift1, 64);
    gemm_bn_kernel<64, 32, true, false, true, true>
        <<<(N + 63) / 64, 128, 0, stream>>>(x, scale1, shift1, W1, w, h1, N);

    // --- CSR build + gather aggregation (replaces 205M fp32 atomics)
    deg_kernel<<<(E + 255) / 256, 256, 0, stream>>>(head, deg, E);
    scan_kernel<<<1, 256, 0, stream>>>(deg, off, N);
    copy_kernel<<<(N + 255) / 256, 256, 0, stream>>>(off, cur, N);
    scatter_kernel<<<(E + 255) / 256, 256, 0, stream>>>(head, cur, perm, E);
    gather_kernel<<<(N + 7) / 8, 256, 0, stream>>>(
        h1, (const float4*)pf, tail, off, perm, (float4*)agg, N);

    // --- BN2 + stage-2 GEMM (in-place over agg): h2 = relu(bn2(agg)) @ W2^T
    hipMemsetAsync(jet, 0, (size_t)B_SEG * 128 * 4, stream);
    jet_kernel<<<(N * 32 + 255) / 256, 256, 0, stream>>>(agg, w, ind, jet, N, 128);
    bnstats_kernel<<<1, 128, 0, stream>>>(jet, g2, b2, scale2, shift2, 128);
    gemm_bn_kernel<128, 128, true, false, true, false>
        <<<(N + 63) / 64, 128, 0, stream>>>(agg, scale2, shift2, W2, nullptr, agg, N);

    // --- Shortcut GEMM straight into d_out: out = x @ Wsc^T
    gemm_bn_kernel<64, 128, false, false, false, false>
        <<<(N + 63) / 64, 128, 0, stream>>>(x, nullptr, nullptr, Wsc, nullptr, out, N);

    // --- BN3 + stage-3 GEMM accumulating onto shortcut (acc init from d_out)
    hipMemsetAsync(jet, 0, (size_t)B_SEG * 128 * 4, stream);
    jet_kernel<<<(N * 32 + 255) / 256, 256, 0, stream>>>(agg, w, ind, jet, N, 128);
    bnstats_kernel<<<1, 128, 0, stream>>>(jet, g3, b3, scale3, shift3, 128);
    gemm_bn_kernel<128, 128, true, true, true, false>
        <<<(N + 63) / 64, 128, 0, stream>>>(agg, scale3, shift3, W3, nullptr, out, N);
}